// MSSTAN_81801947120142
// MI455X (gfx1250) — compile-verified
//
#include <hip/hip_runtime.h>
#include <math.h>

typedef __attribute__((ext_vector_type(2))) float v2f;
typedef __attribute__((ext_vector_type(8))) float v8f;

#define NB    8
#define TG    128
#define NN    90
#define DD    90
#define DK    15
#define NHD   6
#define D2    180
#define DH    30
#define NHSH  8
#define BT    (NB*TG)        /* 1024 graphs (incl. pad slice) */
#define ROWS  (BT*NN)        /* 92160 token rows */
#define GSZ   (NN*NN)        /* 8100 */

// ---------------------------------------------------------------- utilities
__device__ __forceinline__ float dev_gelu(float x) {
  const float c = 0.7978845608028654f; // sqrt(2/pi)
  float t = c * (x + 0.044715f * x * x * x);
  return 0.5f * x * (1.0f + tanhf(t));
}

__device__ __forceinline__ unsigned rotl32(unsigned x, int r) {
  return (x << r) | (x >> (32 - r));
}

// jax threefry2x32 (20 rounds)
__device__ __forceinline__ void threefry2x32(unsigned k0, unsigned k1,
                                             unsigned& x0, unsigned& x1) {
  unsigned ks0 = k0, ks1 = k1, ks2 = k0 ^ k1 ^ 0x1BD11BDAu;
  const int rotA[4] = {13, 15, 26, 6};
  const int rotB[4] = {17, 29, 16, 24};
  x0 += ks0; x1 += ks1;
  unsigned ks[3] = {ks0, ks1, ks2};
  #pragma unroll
  for (int i = 0; i < 5; ++i) {
    const int* rr = (i & 1) ? rotB : rotA;
    #pragma unroll
    for (int r = 0; r < 4; ++r) { x0 += x1; x1 = rotl32(x1, rr[r]); x1 ^= x0; }
    x0 += ks[(i + 1) % 3];
    x1 += ks[(i + 2) % 3] + (unsigned)(i + 1);
  }
}

// Giles single-precision erfinv (matches XLA ErfInv f32)
__device__ __forceinline__ float erfinv_f(float x) {
  float w = -logf((1.0f - x) * (1.0f + x));
  float p;
  if (w < 5.0f) {
    w = w - 2.5f;
    p = 2.81022636e-08f;
    p = 3.43273939e-07f + p * w;
    p = -3.5233877e-06f + p * w;
    p = -4.39150654e-06f + p * w;
    p = 0.00021858087f + p * w;
    p = -0.00125372503f + p * w;
    p = -0.00417768164f + p * w;
    p = 0.246640727f + p * w;
    p = 1.50140941f + p * w;
  } else {
    w = sqrtf(w) - 3.0f;
    p = -0.000200214257f;
    p = 0.000100950558f + p * w;
    p = 0.00134934322f + p * w;
    p = -0.00367342844f + p * w;
    p = 0.00573950773f + p * w;
    p = -0.0076224613f + p * w;
    p = 0.00943887047f + p * w;
    p = 1.00167406f + p * w;
    p = 2.83297682f + p * w;
  }
  return p * x;
}

// jax.random.normal bit pipeline: bits -> [1,2) -> [0,1) -> [lo,1) -> sqrt(2)*erfinv
__device__ __forceinline__ float bits_to_normal(unsigned bits) {
  unsigned fb = 0x3f800000u | (bits >> 9);
  float f = __uint_as_float(fb) - 1.0f;
  const float lo = -0.99999994f;  // nextafter(-1, 0)
  float u = f * (1.0f - lo) + lo;
  u = fmaxf(u, lo);
  return 1.41421356237309515f * erfinv_f(u);
}

// ---------------------------------------------------------------- pad kernel
// src [8][127][8100] -> dst [8][128][8100], last time slice zero
__global__ void k_build_padded(const float* __restrict__ src, float* __restrict__ dst) {
  size_t i = (size_t)blockIdx.x * blockDim.x + threadIdx.x;
  const size_t total = (size_t)BT * GSZ;
  if (i >= total) return;
  size_t g = i / GSZ, e = i % GSZ;
  size_t q = g / TG, t = g % TG;
  dst[i] = (t < TG - 1) ? src[(q * (TG - 1) + t) * GSZ + e] : 0.0f;
}

__global__ void k_zero(float* __restrict__ p, int n) {
  int i = blockIdx.x * blockDim.x + threadIdx.x;
  if (i < n) p[i] = 0.0f;
}

// ---------------------------------------------------------------- WMMA GEMM
// Y[M,N] = act( X[M,K] @ W[N,K]^T + bias[N] )  (+= if ACCUM)
// One wave (32 lanes) per 16x16 output tile, V_WMMA_F32_16X16X4_F32 over K.
// Requires M % 16 == 0.  Main K loop is unguarded (pair loads -> b64);
// only the final partial K-step zeroes the A fragment (zero-A kills the
// products, so B never needs value masking).  Out-of-range N lanes clamp
// their B row pointer in-bounds: the garbage column is never stored.
template <int ACT, bool ACCUM>
__global__ void __launch_bounds__(32)
k_gemm_wmma(const float* __restrict__ X, const float* __restrict__ Wt,
            const float* __restrict__ bias, float* __restrict__ Y,
            int M, int N, int K) {
  int tiles_n = (N + 15) >> 4;
  int tn = blockIdx.x % tiles_n;
  int tm = blockIdx.x / tiles_n;
  int m0 = tm << 4, n0 = tn << 4;
  int lane = threadIdx.x;          // 0..31
  int l15 = lane & 15;
  int khalf = (lane >> 4) << 1;    // 0 or 2
  int arow = m0 + l15;             // always < M (M % 16 == 0)
  int bcol = n0 + l15;
  bool bvalid = bcol < N;
  int bcolc = bvalid ? bcol : 0;
  const float* xp = X + (size_t)arow * K + khalf;
  const float* wp = Wt + (size_t)bcolc * K + khalf;
  int Kmain = K & ~3;
  v8f acc = {};
  for (int k0 = 0; k0 < Kmain; k0 += 4) {
    v2f a = {xp[k0], xp[k0 + 1]};
    v2f b = {wp[k0], wp[k0 + 1]};
    acc = __builtin_amdgcn_wmma_f32_16x16x4_f32(false, a, false, b,
                                                (short)0, acc, false, false);
  }
  if (Kmain < K) {
    int ka = Kmain + khalf;
    int kb = ka + 1;
    int kac = (ka < K) ? ka : 0;       // clamp B column index in-bounds
    int kbc = (kb < K) ? kb : 0;
    const float* xr = X + (size_t)arow * K;
    const float* wr = Wt + (size_t)bcolc * K;
    float a0 = (ka < K) ? xr[kac] : 0.0f;
    float a1 = (kb < K) ? xr[kbc] : 0.0f;
    v2f a = {a0, a1};
    v2f b = {wr[kac], wr[kbc]};
    acc = __builtin_amdgcn_wmma_f32_16x16x4_f32(false, a, false, b,
                                                (short)0, acc, false, false);
  }
  if (bvalid) {
    float bv = bias ? bias[bcol] : 0.0f;
    int rbase = m0 + ((lane >> 4) ? 8 : 0);
    #pragma unroll
    for (int r = 0; r < 8; ++r) {
      float val = acc[r] + bv;
      if (ACT == 1) val = dev_gelu(val);
      size_t idx = (size_t)(rbase + r) * N + bcol;
      if (ACCUM) Y[idx] += val; else Y[idx] = val;
    }
  }
}

// ------------------------------------------------- spatial masked attention
// one block per (graph, head); 96 threads; online softmax over 90 keys
__global__ void k_spatial_attn(const float* __restrict__ qg, const float* __restrict__ kg,
                               const float* __restrict__ vg, const float* __restrict__ ag,
                               float* __restrict__ out) {
  int gh = blockIdx.x;
  int g = gh / NHD, h = gh % NHD;
  __shared__ float ksh[NN * DK];
  __shared__ float vsh[NN * DK];
  int tid = threadIdx.x;
  for (int idx = tid; idx < NN * DK; idx += 96) {
    int n = idx / DK, d = idx % DK;
    size_t base = ((size_t)g * NN + n) * DD + h * DK + d;
    ksh[idx] = kg[base];
    vsh[idx] = vg[base];
  }
  __syncthreads();
  if (tid < NN) {
    float qr[DK];
    const float* qrow = qg + ((size_t)g * NN + tid) * DD + h * DK;
    #pragma unroll
    for (int d = 0; d < DK; ++d) qr[d] = qrow[d];
    const float* arow = ag + (size_t)g * GSZ + (size_t)tid * NN;
    const float scale = 0.25819888974716115f; // 1/sqrt(15)
    float m = -INFINITY, l = 0.0f;
    float acc[DK];
    #pragma unroll
    for (int d = 0; d < DK; ++d) acc[d] = 0.0f;
    for (int j = 0; j < NN; ++j) {
      float dot = 0.0f;
      #pragma unroll
      for (int d = 0; d < DK; ++d) dot += qr[d] * ksh[j * DK + d];
      float s = dot * scale + arow[j];
      if (s > m) {
        float r = expf(m - s);
        l *= r;
        #pragma unroll
        for (int d = 0; d < DK; ++d) acc[d] *= r;
        m = s;
      }
      float p = expf(s - m);
      l += p;
      #pragma unroll
      for (int d = 0; d < DK; ++d) acc[d] += p * vsh[j * DK + d];
    }
    float inv = 1.0f / l;
    float* orow = out + ((size_t)g * NN + tid) * DD + h * DK;
    #pragma unroll
    for (int d = 0; d < DK; ++d) orow[d] = acc[d] * inv;
  }
}

// ------------------------------------------------------- y = x + LN(x) rows
// torch-style: unbiased var (ddof=1), eps added to std
__global__ void k_add_ln(const float* __restrict__ x, const float* __restrict__ a,
                         const float* __restrict__ b, float* __restrict__ y,
                         int rows, int D) {
  int r = blockIdx.x * blockDim.x + threadIdx.x;
  if (r >= rows) return;
  const float* xr = x + (size_t)r * D;
  float mean = 0.0f;
  for (int d = 0; d < D; ++d) mean += xr[d];
  mean /= (float)D;
  float var = 0.0f;
  for (int d = 0; d < D; ++d) { float t = xr[d] - mean; var += t * t; }
  var /= (float)(D - 1);
  float inv = 1.0f / (sqrtf(var) + 1e-6f);
  float* yr = y + (size_t)r * D;
  for (int d = 0; d < D; ++d) {
    float xv = xr[d];
    yr[d] = xv + a[d] * (xv - mean) * inv + b[d];
  }
}

// ------------------------------------------------- self-attention topk pool
// block per (pool, graph); 128 threads
__global__ void k_pool_topk(const float* __restrict__ h, const float* __restrict__ ag,
                            const float* __restrict__ pw, float* __restrict__ gs) {
  int p = blockIdx.x / BT;
  int g = blockIdx.x % BT;
  __shared__ float tvec[NN];
  __shared__ float score[NN];
  __shared__ int topi[64];
  __shared__ float topv[64];
  int tid = threadIdx.x;
  if (tid < NN) {
    float s = 0.0f;
    const float* hr = h + ((size_t)g * NN + tid) * DD;
    for (int d = 0; d < DD; ++d) s += hr[d] * pw[p * DD + d];
    tvec[tid] = s;
  }
  __syncthreads();
  if (tid < NN) {
    float s = 0.0f;
    const float* arow = ag + (size_t)g * GSZ + (size_t)tid * NN;
    for (int j = 0; j < NN; ++j) s += arow[j] * tvec[j];
    score[tid] = s;
  }
  __syncthreads();
  int kk = (p == 0) ? 27 : (p == 1) ? 45 : 63;   // ceil(r*N)
  if (tid == 0) {
    bool used[NN];
    for (int n = 0; n < NN; ++n) used[n] = false;
    for (int sel = 0; sel < kk; ++sel) {
      float best = -INFINITY; int bi = 0;
      for (int n = 0; n < NN; ++n)
        if (!used[n] && score[n] > best) { best = score[n]; bi = n; }
      used[bi] = true;
      topi[sel] = bi;
      topv[sel] = tanhf(best);
    }
  }
  __syncthreads();
  for (int d = tid; d < DD; d += blockDim.x) {
    float mn = 0.0f, mx = -INFINITY;
    for (int s = 0; s < kk; ++s) {
      float val = h[((size_t)g * NN + topi[s]) * DD + d] * topv[s];
      mn += val;
      mx = fmaxf(mx, val);
    }
    float* go = gs + ((size_t)p * BT + g) * D2;
    go[d] = mn / (float)kk;
    go[DD + d] = mx;
  }
}

// -------------------------------------------- LSH round: rotation generation
// reproduces jax.random.normal(jax.random.key(seed), (30,8,1)) via threefry2x32
__global__ void k_lsh_rot(unsigned seed, float* __restrict__ rot) {
  int j = threadIdx.x;
  if (j >= 120) return;
  unsigned x0 = (unsigned)j, x1 = (unsigned)(j + 120);
  threefry2x32(0u, seed, x0, x1);
  rot[j] = bits_to_normal(x0);
  rot[j + 120] = bits_to_normal(x1);
}

// -------------------------------------------- LSH bucket + counting argsort
// block per batch-head; bt = bucket*128 + t values are unique and < 2048
__global__ void k_lsh_sort(const float* __restrict__ qk, const float* __restrict__ rot,
                           int* __restrict__ stg, int* __restrict__ undog) {
  int bh = blockIdx.x;
  int q = bh / NHD, hd = bh % NHD;
  __shared__ unsigned char cnt[2048];
  __shared__ short idxof[2048];
  int tid = threadIdx.x;
  for (int b = tid; b < 2048; b += blockDim.x) cnt[b] = 0;
  __syncthreads();
  for (int i = tid; i < NHSH * TG; i += blockDim.x) {
    int h = i >> 7, t = i & 127;
    const float* row = qk + ((size_t)(q * TG + t)) * D2 + hd * DH;
    float s = 0.0f;
    for (int d = 0; d < DH; ++d) s += row[d] * rot[d * NHSH + h];
    int bucket = (s < 0.0f) ? 1 : 0;           // argmax([x,-x]) tie -> 0
    int bt = ((h * 2 + bucket) << 7) | t;
    cnt[bt] = 1;
    idxof[bt] = (short)i;
  }
  __syncthreads();
  if (tid == 0) {
    int pos = 0;
    for (int b = 0; b < 2048; ++b) {
      if (cnt[b]) {
        int i = idxof[b];
        stg[bh * (NHSH * TG) + pos] = i & 127;   // sorted token position
        undog[bh * (NHSH * TG) + i] = pos;
        pos++;
      }
    }
  }
}

// -------------------------------------------- LSH chunked attention (64x128)
// block per (batch-head, chunk); 64 threads (one query row each)
__global__ void k_lsh_chunk(const float* __restrict__ qk, const float* __restrict__ vv,
                            const int* __restrict__ stg, float* __restrict__ sbo,
                            float* __restrict__ slog) {
  int bh = blockIdx.x >> 4;
  int c = blockIdx.x & 15;
  int q = bh / NHD, hd = bh % NHD;
  __shared__ float ks[128][DH];
  __shared__ float vs[128][DH];
  __shared__ float kinv[128];
  __shared__ int tpos[128];
  int tid = threadIdx.x;
  for (int j = tid; j < 128; j += 64) {
    int cc = (j < 64) ? c : (c + 15) & 15;   // look-back one chunk (roll 1)
    int spos = cc * 64 + (j & 63);
    int tok = stg[bh * (NHSH * TG) + spos];
    tpos[j] = tok;
    const float* row = qk + ((size_t)(q * TG + tok)) * D2 + hd * DH;
    const float* vrow = vv + ((size_t)(q * TG + tok)) * D2 + hd * DH;
    float nrm = 0.0f;
    for (int d = 0; d < DH; ++d) {
      float val = row[d];
      ks[j][d] = val;
      nrm += val * val;
      vs[j][d] = vrow[d];
    }
    kinv[j] = 1.0f / sqrtf(nrm);  // keys unit-normalized (0-rows -> NaN like ref)
  }
  __syncthreads();
  const float scale = 0.18257418583505536f; // 1/sqrt(30)
  int i = tid;
  int myt = tpos[i];
  float m = -INFINITY, l = 0.0f;
  float acc[DH];
  #pragma unroll
  for (int d = 0; d < DH; ++d) acc[d] = 0.0f;
  for (int j = 0; j < 128; ++j) {
    float dot = 0.0f;
    #pragma unroll
    for (int d = 0; d < DH; ++d) dot += ks[i][d] * ks[j][d];
    float s = dot * kinv[j] * scale;
    if (tpos[j] == myt) s = -5e4f;           // reformer self-mask
    if (s > m) {
      float r = expf(m - s);
      l *= r;
      #pragma unroll
      for (int d = 0; d < DH; ++d) acc[d] *= r;
      m = s;
    }
    float p = expf(s - m);
    l += p;
    #pragma unroll
    for (int d = 0; d < DH; ++d) acc[d] += p * vs[j][d];
  }
  float inv = 1.0f / l;
  size_t obase = ((size_t)bh * (NHSH * TG) + c * 64 + i);
  float* orow = sbo + obase * DH;
  #pragma unroll
  for (int d = 0; d < DH; ++d) orow[d] = acc[d] * inv;
  slog[obase] = m + logf(l);
}

// ----------------------------------- undo sort + softmax over 8 hash rounds
__global__ void k_lsh_combine(const float* __restrict__ sbo, const float* __restrict__ slog,
                              const int* __restrict__ undog, float* __restrict__ attn) {
  int bh = blockIdx.x;
  int t = threadIdx.x;  // 0..127
  int q = bh / NHD, hd = bh % NHD;
  int pos[NHSH];
  float lg[NHSH];
  float m = -INFINITY;
  #pragma unroll
  for (int h = 0; h < NHSH; ++h) {
    int p = undog[bh * (NHSH * TG) + h * TG + t];
    pos[h] = p;
    lg[h] = slog[bh * (NHSH * TG) + p];
    m = fmaxf(m, lg[h]);
  }
  float w[NHSH];
  float denom = 0.0f;
  #pragma unroll
  for (int h = 0; h < NHSH; ++h) { w[h] = expf(lg[h] - m); denom += w[h]; }
  float inv = 1.0f / denom;
  float* orow = attn + ((size_t)(q * TG + t)) * D2 + hd * DH;
  for (int d = 0; d < DH; ++d) {
    float s = 0.0f;
    #pragma unroll
    for (int h = 0; h < NHSH; ++h)
      s += w[h] * sbo[((size_t)bh * (NHSH * TG) + pos[h]) * DH + d];
    orow[d] = s * inv;
  }
}

// ----------------------------------------------- classifier head (one block)
__global__ void k_classifier(const float* __restrict__ outT,
                             const float* __restrict__ fc1W, const float* __restrict__ fc1b,
                             const float* __restrict__ fc2W, const float* __restrict__ fc2b,
                             const float* __restrict__ fc3W, const float* __restrict__ fc3b,
                             float* __restrict__ out) {
  __shared__ float z[NB * D2];
  __shared__ float h1[NB * 128];
  __shared__ float h2[NB * 64];
  int tid = threadIdx.x, nt = blockDim.x;
  for (int idx = tid; idx < NB * D2; idx += nt) {
    int q = idx / D2, d = idx % D2;
    float s = 0.0f;
    for (int t = 0; t < TG; ++t) s += outT[((size_t)(q * TG + t)) * D2 + d];
    z[idx] = s / (float)TG;
  }
  __syncthreads();
  for (int idx = tid; idx < NB * 128; idx += nt) {
    int q = idx / 128, o = idx % 128;
    float s = fc1b[o];
    for (int d = 0; d < D2; ++d) s += z[q * D2 + d] * fc1W[o * D2 + d];
    h1[idx] = fmaxf(s, 0.0f);
  }
  __syncthreads();
  for (int idx = tid; idx < NB * 64; idx += nt) {
    int q = idx / 64, o = idx % 64;
    float s = fc2b[o];
    for (int d = 0; d < 128; ++d) s += h1[q * 128 + d] * fc2W[o * 128 + d];
    h2[idx] = fmaxf(s, 0.0f);
  }
  __syncthreads();
  if (tid < NB) {
    float l0 = fc3b[0], l1 = fc3b[1];
    for (int d = 0; d < 64; ++d) {
      l0 += h2[tid * 64 + d] * fc3W[d];
      l1 += h2[tid * 64 + d] * fc3W[64 + d];
    }
    float mm = fmaxf(l0, l1);
    float lse = mm + logf(expf(l0 - mm) + expf(l1 - mm));
    out[tid * 2 + 0] = l0 - lse;
    out[tid * 2 + 1] = l1 - lse;
  }
}

// ---------------------------------------------------------------- launcher
extern "C" void kernel_launch(void* const* d_in, const int* in_sizes, int n_in,
                              void* d_out, int out_size, void* d_ws, size_t ws_size,
                              hipStream_t stream) {
  const float* x    = (const float*)d_in[0];
  const float* adj  = (const float*)d_in[1];
  const float* sWq  = (const float*)d_in[2];
  const float* sbq  = (const float*)d_in[3];
  const float* sWk  = (const float*)d_in[4];
  const float* sbk  = (const float*)d_in[5];
  const float* sWv  = (const float*)d_in[6];
  const float* sbv  = (const float*)d_in[7];
  const float* sWo  = (const float*)d_in[8];
  const float* sbo_in = (const float*)d_in[9];
  const float* n1a  = (const float*)d_in[10];
  const float* n1b  = (const float*)d_in[11];
  const float* n2a  = (const float*)d_in[12];
  const float* n2b  = (const float*)d_in[13];
  const float* fw1  = (const float*)d_in[14];
  const float* fb1  = (const float*)d_in[15];
  const float* fw2  = (const float*)d_in[16];
  const float* fb2  = (const float*)d_in[17];
  const float* pw   = (const float*)d_in[18];
  const float* tWqk = (const float*)d_in[19];
  const float* tWv  = (const float*)d_in[20];
  const float* tWo  = (const float*)d_in[21];
  const float* tbo  = (const float*)d_in[22];
  const float* fc1W = (const float*)d_in[23];
  const float* fc1b = (const float*)d_in[24];
  const float* fc2W = (const float*)d_in[25];
  const float* fc2b = (const float*)d_in[26];
  const float* fc3W = (const float*)d_in[27];
  const float* fc3b = (const float*)d_in[28];
  (void)in_sizes; (void)n_in; (void)out_size; (void)ws_size;

  // ---- workspace layout (floats) ----
  const size_t S = (size_t)BT * GSZ;       // 8,294,400
  float* W0  = (float*)d_ws;
  float* xg  = W0 + 0 * S;
  float* agb = W0 + 1 * S;
  float* qb  = W0 + 2 * S;
  float* kb  = W0 + 3 * S;
  float* vb  = W0 + 4 * S;
  float* h1  = W0 + 5 * S;
  float* hA  = qb;        // reuse (q dead after attention)
  float* hn  = kb;        // reuse (k dead)
  float* mid = vb;        // 2S region spanning vb..h1 (both dead)
  float* h2b = qb;        // reuse
  float* hf  = xg;        // reuse (xg dead)
  float* sm   = W0 + 6 * S;
  float* gsb  = sm;                                // 3*1024*180
  float* qkb  = gsb  + (size_t)3 * BT * D2;        // 1024*180
  float* vvb  = qkb  + (size_t)BT * D2;
  float* rotb = vvb  + (size_t)BT * D2;            // 240 (reserve 256)
  float* sboB = rotb + 256;                        // 48*1024*30
  float* slgB = sboB + (size_t)48 * 1024 * DH;     // 48*1024
  float* atnB = slgB + (size_t)48 * 1024;          // 1024*180
  float* outT = atnB + (size_t)BT * D2;            // 1024*180
  int*   stg  = (int*)(outT + (size_t)BT * D2);    // 48*1024
  int*   und  = stg + 48 * 1024;                   // 48*1024

  const size_t padN = (size_t)BT * GSZ;
  int padBlocks = (int)((padN + 255) / 256);
  k_build_padded<<<padBlocks, 256, 0, stream>>>(x, xg);
  k_build_padded<<<padBlocks, 256, 0, stream>>>(adj, agb);

  // ---- spatial transformer ----
  {
    int tm = ROWS / 16, tn = (DD + 15) / 16;   // 5760 x 6
    dim3 grid(tm * tn);
    k_gemm_wmma<0, false><<<grid, 32, 0, stream>>>(xg, sWq, sbq, qb, ROWS, DD, DD);
    k_gemm_wmma<0, false><<<grid, 32, 0, stream>>>(xg, sWk, sbk, kb, ROWS, DD, DD);
    k_gemm_wmma<0, false><<<grid, 32, 0, stream>>>(xg, sWv, sbv, vb, ROWS, DD, DD);
  }
  k_spatial_attn<<<BT * NHD, 96, 0, stream>>>(qb, kb, vb, agb, h1);
  {
    int tm = ROWS / 16, tn = (DD + 15) / 16;
    k_gemm_wmma<0, false><<<dim3(tm * tn), 32, 0, stream>>>(h1, sWo, sbo_in, hA, ROWS, DD, DD);
  }
  k_add_ln<<<(ROWS + 255) / 256, 256, 0, stream>>>(hA, n1a, n1b, hn, ROWS, DD);
  {
    int tm = ROWS / 16;
    k_gemm_wmma<1, false><<<dim3(tm * ((D2 + 15) / 16)), 32, 0, stream>>>(hn, fw1, fb1, mid, ROWS, D2, DD);
    k_gemm_wmma<0, false><<<dim3(tm * ((DD + 15) / 16)), 32, 0, stream>>>(mid, fw2, fb2, h2b, ROWS, DD, D2);
  }
  k_add_ln<<<(ROWS + 255) / 256, 256, 0, stream>>>(h2b, n2a, n2b, hf, ROWS, DD);

  // ---- top-k pooling x3 ----
  k_pool_topk<<<3 * BT, 128, 0, stream>>>(hf, agb, pw, gsb);

  // ---- 3x LSH temporal attention, summed ----
  k_zero<<<(BT * D2 + 255) / 256, 256, 0, stream>>>(outT, BT * D2);
  for (int i = 0; i < 3; ++i) {
    const float* gsi = gsb + (size_t)i * BT * D2;
    k_lsh_rot<<<1, 128, 0, stream>>>((unsigned)(42 + i), rotb);
    {
      int tm = BT / 16, tn = (D2 + 15) / 16;   // 64 x 12
      dim3 grid(tm * tn);
      k_gemm_wmma<0, false><<<grid, 32, 0, stream>>>(gsi, tWqk + (size_t)i * D2 * D2,
                                                     nullptr, qkb, BT, D2, D2);
      k_gemm_wmma<0, false><<<grid, 32, 0, stream>>>(gsi, tWv + (size_t)i * D2 * D2,
                                                     nullptr, vvb, BT, D2, D2);
    }
    k_lsh_sort<<<48, 256, 0, stream>>>(qkb, rotb, stg, und);
    k_lsh_chunk<<<48 * 16, 64, 0, stream>>>(qkb, vvb, stg, sboB, slgB);
    k_lsh_combine<<<48, 128, 0, stream>>>(sboB, slgB, und, atnB);
    {
      int tm = BT / 16, tn = (D2 + 15) / 16;
      k_gemm_wmma<0, true><<<dim3(tm * tn), 32, 0, stream>>>(atnB, tWo + (size_t)i * D2 * D2,
                                                             tbo + (size_t)i * D2, outT,
                                                             BT, D2, D2);
    }
  }

  // ---- classifier head ----
  k_classifier<<<1, 512, 0, stream>>>(outT, fc1W, fc1b, fc2W, fc2b, fc3W, fc3b,
                                      (float*)d_out);
}